// NetLinkEvaluate_86234353369871
// MI455X (gfx1250) — compile-verified
//
#include <hip/hip_runtime.h>

typedef float v2f __attribute__((ext_vector_type(2)));
typedef float v8f __attribute__((ext_vector_type(8)));

#define NFEAT 128
#define NHID  64

// ---------------------------------------------------------------------------
// Dense transform: C[nrows,64] = (RELU? max(A,0) : A)[nrows,K] @ B[K,64]
// One wave computes a 16x64 strip using V_WMMA_F32_16X16X4_F32.
// B is staged transposed in LDS (Bs[n*K + k]) so each lane's K-pair for the
// B fragment is a single 8-byte ds_load.
// Fragment layout (16x16x4 f32, wave32):
//   A: lane<16 -> {K=k0,k0+1}, lane>=16 -> {K=k0+2,k0+3}; M = lane%16
//   B: same K split; N = lane%16
//   C/D: vgpr r -> row (r + 8*half), col = lane%16
// ---------------------------------------------------------------------------
template <int K, bool RELU>
__global__ __launch_bounds__(256) void gemm_wmma_kernel(
    const float* __restrict__ A, const float* __restrict__ B,
    float* __restrict__ C, int nrows) {
  __shared__ float Bs[64 * K];  // transposed: Bs[n*K + k]
  for (int i = threadIdx.x; i < K * 64; i += 256) {
    int k = i / 64, n = i % 64;
    Bs[n * K + k] = B[i];
  }
  __syncthreads();

  const int wave = threadIdx.x >> 5;
  const int lane = threadIdx.x & 31;
  const int idx  = lane & 15;
  const int half = lane >> 4;

  const int strip = blockIdx.x * 8 + wave;  // 8 waves / block
  const int r0 = strip * 16;
  if (r0 >= nrows) return;  // wave-uniform: EXEC stays all-ones for WMMA

  int arow = r0 + idx;
  if (arow >= nrows) arow = nrows - 1;  // clamp (keeps EXEC full)
  const float* __restrict__ Arow = A + (size_t)arow * K;

  v8f acc0 = {}, acc1 = {}, acc2 = {}, acc3 = {};

#pragma unroll
  for (int k0 = 0; k0 < K; k0 += 4) {
    const int ka = k0 + 2 * half;  // even -> 8B aligned
    v2f a = *(const v2f*)&Arow[ka];
    if (RELU) {
      a.x = fmaxf(a.x, 0.0f);
      a.y = fmaxf(a.y, 0.0f);
    }
    const v2f b0 = *(const v2f*)&Bs[(idx +  0) * K + ka];
    const v2f b1 = *(const v2f*)&Bs[(idx + 16) * K + ka];
    const v2f b2 = *(const v2f*)&Bs[(idx + 32) * K + ka];
    const v2f b3 = *(const v2f*)&Bs[(idx + 48) * K + ka];
    acc0 = __builtin_amdgcn_wmma_f32_16x16x4_f32(false, a, false, b0, (short)0, acc0, false, false);
    acc1 = __builtin_amdgcn_wmma_f32_16x16x4_f32(false, a, false, b1, (short)0, acc1, false, false);
    acc2 = __builtin_amdgcn_wmma_f32_16x16x4_f32(false, a, false, b2, (short)0, acc2, false, false);
    acc3 = __builtin_amdgcn_wmma_f32_16x16x4_f32(false, a, false, b3, (short)0, acc3, false, false);
  }

  const int rbase = r0 + 8 * half;
#pragma unroll
  for (int r = 0; r < 8; ++r) {
    const int row = rbase + r;
    if (row < nrows) {
      float* __restrict__ Crow = C + (size_t)row * 64;
      Crow[idx +  0] = acc0[r];
      Crow[idx + 16] = acc1[r];
      Crow[idx + 32] = acc2[r];
      Crow[idx + 48] = acc3[r];
    }
  }
}

// ---------------------------------------------------------------------------
// Edge message passing: OUT[dst[e],:] += w[e] * XW[src[e],:]
// 16 threads per edge, float4 per thread, hardware f32 global atomics.
// ---------------------------------------------------------------------------
__global__ __launch_bounds__(256) void scatter_kernel(
    const float* __restrict__ XW, const int* __restrict__ src,
    const int* __restrict__ dst, const float* __restrict__ w,
    float* __restrict__ OUT, int nedges) {
  const long long gid = (long long)blockIdx.x * 256 + threadIdx.x;
  const int e = (int)(gid >> 4);
  const int q = (int)(gid & 15);
  if (e >= nedges) return;
  const int s = src[e], d = dst[e];
  const float wt = w[e];
  const float4 v = *(const float4*)(XW + (size_t)s * 64 + q * 4);
  float* o = OUT + (size_t)d * 64 + q * 4;
  __hip_atomic_fetch_add(o + 0, v.x * wt, __ATOMIC_RELAXED, __HIP_MEMORY_SCOPE_AGENT);
  __hip_atomic_fetch_add(o + 1, v.y * wt, __ATOMIC_RELAXED, __HIP_MEMORY_SCOPE_AGENT);
  __hip_atomic_fetch_add(o + 2, v.z * wt, __ATOMIC_RELAXED, __HIP_MEMORY_SCOPE_AGENT);
  __hip_atomic_fetch_add(o + 3, v.w * wt, __ATOMIC_RELAXED, __HIP_MEMORY_SCOPE_AGENT);
}

// ---------------------------------------------------------------------------
// Link decode: out[p,:2] = [z[s]||z[d]] @ Wdec^T, Wdec is [2,128] row-major.
// ---------------------------------------------------------------------------
__global__ __launch_bounds__(256) void decode_kernel(
    const float* __restrict__ Z, const int* __restrict__ psrc,
    const int* __restrict__ pdst, const float* __restrict__ Wdec,
    float* __restrict__ out, int npe) {
  __shared__ float Wd[256];
  Wd[threadIdx.x] = Wdec[threadIdx.x];
  __syncthreads();
  const int p = blockIdx.x * 256 + threadIdx.x;
  if (p >= npe) return;
  const int s = psrc[p], d = pdst[p];
  const float4* zs = (const float4*)(Z + (size_t)s * 64);
  const float4* zd = (const float4*)(Z + (size_t)d * 64);
  float a0 = 0.0f, a1 = 0.0f;
#pragma unroll
  for (int c = 0; c < 16; ++c) {
    const float4 v = zs[c];
    const int b = 4 * c;
    a0 += v.x * Wd[b] + v.y * Wd[b + 1] + v.z * Wd[b + 2] + v.w * Wd[b + 3];
    a1 += v.x * Wd[128 + b] + v.y * Wd[129 + b] + v.z * Wd[130 + b] + v.w * Wd[131 + b];
    const float4 u = zd[c];
    a0 += u.x * Wd[64 + b] + u.y * Wd[65 + b] + u.z * Wd[66 + b] + u.w * Wd[67 + b];
    a1 += u.x * Wd[192 + b] + u.y * Wd[193 + b] + u.z * Wd[194 + b] + u.w * Wd[195 + b];
  }
  out[(size_t)p * 2 + 0] = a0;
  out[(size_t)p * 2 + 1] = a1;
}

extern "C" void kernel_launch(void* const* d_in, const int* in_sizes, int n_in,
                              void* d_out, int out_size, void* d_ws, size_t ws_size,
                              hipStream_t stream) {
  (void)n_in; (void)out_size; (void)ws_size;
  const float* x    = (const float*)d_in[0];  // [N,128]
  const int*   ei   = (const int*)d_in[1];    // [2,E]
  const float* ew   = (const float*)d_in[2];  // [E]
  const int*   pei  = (const int*)d_in[3];    // [2,PE]
  const float* W1   = (const float*)d_in[4];  // [128,64]
  const float* W2   = (const float*)d_in[5];  // [64,64]
  const float* Wdec = (const float*)d_in[6];  // [2,128]
  float* out = (float*)d_out;                 // [PE,2]

  const int N  = in_sizes[0] / NFEAT;
  const int E  = in_sizes[2];
  const int PE = in_sizes[3] / 2;

  float* buf0 = (float*)d_ws;                     // [N,64]
  float* buf1 = buf0 + (size_t)N * NHID;          // [N,64]
  const size_t zbytes = (size_t)N * NHID * sizeof(float);

  const int strips     = (N + 15) / 16;
  const int gemm_grid  = (strips + 7) / 8;
  const long long sthreads = (long long)E * 16;
  const int scat_grid  = (int)((sthreads + 255) / 256);
  const int dec_grid   = (PE + 255) / 256;

  // Layer 1: xw1 = x @ W1   -> buf0
  gemm_wmma_kernel<NFEAT, false><<<gemm_grid, 256, 0, stream>>>(x, W1, buf0, N);
  // z1 = scatter_sum(w * xw1[src] -> dst)   -> buf1
  hipMemsetAsync(buf1, 0, zbytes, stream);
  scatter_kernel<<<scat_grid, 256, 0, stream>>>(buf0, ei, ei + E, ew, buf1, E);
  // Layer 2: zw2 = relu(z1) @ W2   -> buf0 (ReLU fused into A load)
  gemm_wmma_kernel<NHID, true><<<gemm_grid, 256, 0, stream>>>(buf1, W2, buf0, N);
  // z2 = scatter_sum(w * zw2[src] -> dst)   -> buf1
  hipMemsetAsync(buf1, 0, zbytes, stream);
  scatter_kernel<<<scat_grid, 256, 0, stream>>>(buf0, ei, ei + E, ew, buf1, E);
  // Decode links
  decode_kernel<<<dec_grid, 256, 0, stream>>>(buf1, pei, pei + PE, Wdec, out, PE);
}